// OpndetBboxLoss_83451214561450
// MI455X (gfx1250) — compile-verified
//
#include <hip/hip_runtime.h>
#include <math.h>

#define B_ 64
#define H_ 192
#define W_ 256
#define HW_ (H_*W_)
#define STRIDE_ 4.0f
#define IMG_W_ 1024.0f
#define IMG_H_ 768.0f
#define KPOS 128
#define EPS_ 1e-7f
#define FOUR_OVER_PI2 0.4052847345693511f

typedef float v2f __attribute__((ext_vector_type(2)));
typedef float v8f __attribute__((ext_vector_type(8)));

__device__ __forceinline__ float sigmoidf_(float x) { return 1.0f / (1.0f + __expf(-x)); }

// ---------------------------------------------------------------------------
// Kernel 1: fused elementwise pass.
//  - focal heatmap loss (pos/neg sums + n_pos)
//  - 5x5 peak suppression via LDS tile w/ halo -> per-image suppressed count
//  - at positives: L1(cxy) and CIoU numerators (sparse gathers)
// Block = 32x8 pixels, LDS sigmoid tile (8+4)x(32+4).
// ---------------------------------------------------------------------------
__global__ __launch_bounds__(256) void fused_loss_kernel(
    const float* __restrict__ raw, const float* __restrict__ tgt_hm,
    const float* __restrict__ tgt_cxy, const float* __restrict__ tgt_wh,
    const float* __restrict__ tgt_pos, float* __restrict__ ws)
{
    __shared__ float s_sig[12][36];
    __shared__ float acc[6];
    const int b = blockIdx.z;
    const int tx = threadIdx.x, ty = threadIdx.y;
    const int tid = ty * 32 + tx;
    const int w0 = blockIdx.x * 32, h0 = blockIdx.y * 8;
    const size_t rbase = (size_t)b * 5 * HW_;

    // cooperative halo load (each heatmap byte read once from HBM)
    for (int t = tid; t < 12 * 36; t += 256) {
        int rh = t / 36, rw = t - rh * 36;
        int gh = h0 - 2 + rh, gw = w0 - 2 + rw;
        float v = -INFINITY;
        if (gh >= 0 && gh < H_ && gw >= 0 && gw < W_)
            v = sigmoidf_(raw[rbase + (size_t)gh * W_ + gw]);
        s_sig[rh][rw] = v;
    }
    if (tid < 6) acc[tid] = 0.0f;
    __syncthreads();

    const int h = h0 + ty, w = w0 + tx;
    const size_t p = (size_t)h * W_ + w;
    const float s = s_sig[ty + 2][tx + 2];

    // 5x5 SAME max pool from LDS
    float pooled = -INFINITY;
#pragma unroll
    for (int dy = 0; dy < 5; ++dy)
#pragma unroll
        for (int dx = 0; dx < 5; ++dx)
            pooled = fmaxf(pooled, s_sig[ty + dy][tx + dx]);
    float msk = (s + 0.005f - pooled) * 200.0f;           // /PEAK_EPS
    msk = fminf(fmaxf(msk, 0.0f), 1.0f);
    const float supp = s * msk;

    // focal heatmap
    const float gt = tgt_hm[(size_t)b * HW_ + p];
    const float pred = fminf(fmaxf(s, 1e-6f), 1.0f - 1e-6f);
    const float posm = (gt == 1.0f) ? 1.0f : 0.0f;
    const float omp = 1.0f - pred;
    const float pl = -(omp * omp) * logf(pred) * posm;
    const float t1 = 1.0f - gt; const float t2 = t1 * t1;
    const float nl = -(pred * pred) * logf(1.0f - pred) * (t2 * t2) * (1.0f - posm);

    // sparse positives: L1 + CIoU
    float cxy_c = 0.0f, ciou_c = 0.0f;
    const float pv = tgt_pos[(size_t)b * HW_ + p];
    if (pv > 0.5f) {
        const size_t cbase = (size_t)b * 2 * HW_;
        float c0  = sigmoidf_(raw[rbase + 1 * (size_t)HW_ + p]);
        float c1  = sigmoidf_(raw[rbase + 2 * (size_t)HW_ + p]);
        float ws_ = sigmoidf_(raw[rbase + 3 * (size_t)HW_ + p]);
        float hs_ = sigmoidf_(raw[rbase + 4 * (size_t)HW_ + p]);
        float tcx = tgt_cxy[cbase + p];
        float tcy = tgt_cxy[cbase + HW_ + p];
        float twd = tgt_wh[cbase + p];
        float tht = tgt_wh[cbase + HW_ + p];
        cxy_c = fabsf(c0 - tcx) + fabsf(c1 - tcy);
        // decode boxes
        float pcx = ((float)w + c0) * STRIDE_, pcy = ((float)h + c1) * STRIDE_;
        float pwp = ws_ * IMG_W_, php = hs_ * IMG_H_;
        float px1 = pcx - pwp * 0.5f, py1 = pcy - php * 0.5f;
        float px2 = pcx + pwp * 0.5f, py2 = pcy + php * 0.5f;
        float gcx = ((float)w + tcx) * STRIDE_, gcy = ((float)h + tcy) * STRIDE_;
        float gwp = twd * IMG_W_, ghp = tht * IMG_H_;
        float gx1 = gcx - gwp * 0.5f, gy1 = gcy - ghp * 0.5f;
        float gx2 = gcx + gwp * 0.5f, gy2 = gcy + ghp * 0.5f;
        float pw = fmaxf(px2 - px1, 0.0f), ph = fmaxf(py2 - py1, 0.0f);
        float gw = fmaxf(gx2 - gx1, 0.0f), gh = fmaxf(gy2 - gy1, 0.0f);
        float iw = fmaxf(fminf(px2, gx2) - fmaxf(px1, gx1), 0.0f);
        float ih = fmaxf(fminf(py2, gy2) - fmaxf(py1, gy1), 0.0f);
        float inter = iw * ih;
        float uni = pw * ph + gw * gh - inter + EPS_;
        float iou = inter / uni;
        float cw = fmaxf(fmaxf(px2, gx2) - fminf(px1, gx1), 0.0f);
        float ch = fmaxf(fmaxf(py2, gy2) - fminf(py1, gy1), 0.0f);
        float cdiag = cw * cw + ch * ch + EPS_;
        float ddx = (px1 + px2) * 0.5f - (gx1 + gx2) * 0.5f;
        float ddy = (py1 + py2) * 0.5f - (gy1 + gy2) * 0.5f;
        float cd2 = ddx * ddx + ddy * ddy;
        float at = atanf(gw / (gh + EPS_)) - atanf(pw / (ph + EPS_));
        float v = FOUR_OVER_PI2 * at * at;
        float alpha = v / (1.0f - iou + v + EPS_);
        ciou_c = 1.0f - iou + cd2 / cdiag + alpha * v;
    }

    atomicAdd(&acc[0], pl);
    atomicAdd(&acc[1], nl);
    atomicAdd(&acc[2], posm);
    atomicAdd(&acc[3], supp);
    atomicAdd(&acc[4], cxy_c);
    atomicAdd(&acc[5], ciou_c);
    __syncthreads();
    if (tid == 0) {
        atomicAdd(&ws[0], acc[0]);       // pos focal sum
        atomicAdd(&ws[1], acc[1]);       // neg focal sum
        atomicAdd(&ws[2], acc[2]);       // n_pos (gt==1)
        atomicAdd(&ws[8 + b], acc[3]);   // suppressed count per image
        atomicAdd(&ws[4], acc[4]);       // cxy numerator
        atomicAdd(&ws[5], acc[5]);       // ciou numerator
    }
}

// ---------------------------------------------------------------------------
// Kernel 2: repulsion. One block (8 waves) per image.
// Pairwise 128x128 d2 via V_WMMA_F32_16X16X4_F32:
//   A_i = [-2cx_i, -2cy_i, r_i, 1],  B_j = [cx_j, cy_j, 1, r_j]^T,
//   (A x B)_ij = r_i + r_j - 2(cx_i cx_j + cy_i cy_j) = d2_ij,  K = 4 exactly.
// Per-row argmin via 16-lane shfl_xor min reduction (C/D layout: row = 8*hi+v,
// col = lane%16 per ISA 7.12.2), then IoA vs the NN's gt box.
// ---------------------------------------------------------------------------
__global__ __launch_bounds__(256) void repulsion_kernel(
    const float* __restrict__ raw, const float* __restrict__ tgt_cxy,
    const float* __restrict__ tgt_wh, const float* __restrict__ tgt_pos,
    float* __restrict__ ws)
{
    __shared__ float s_px[KPOS], s_py[KPOS], s_r[KPOS];
    __shared__ float s_g[4][KPOS];   // gt boxes
    __shared__ float s_m[4][KPOS];   // pred boxes
    __shared__ int   s_nn[KPOS];
    __shared__ int   s_cnt;
    __shared__ float s_red[KPOS];
    const int b = blockIdx.x;
    const int tid = threadIdx.x;
    if (tid == 0) s_cnt = 0;
    __syncthreads();

    const size_t pbase = (size_t)b * HW_;
    const size_t rbase = (size_t)b * 5 * HW_;
    const size_t cbase = (size_t)b * 2 * HW_;

    // collect positives (set semantics: order irrelevant for NN/mean)
    for (int p = tid; p < HW_; p += 256) {
        if (tgt_pos[pbase + p] > 0.5f) {
            int slot = atomicAdd(&s_cnt, 1);
            if (slot < KPOS) {
                int h = p / W_, w = p - h * W_;
                float tcx = tgt_cxy[cbase + p];
                float tcy = tgt_cxy[cbase + HW_ + p];
                float twd = tgt_wh[cbase + p];
                float tht = tgt_wh[cbase + HW_ + p];
                float cx = ((float)w + tcx) * STRIDE_;
                float cy = ((float)h + tcy) * STRIDE_;
                float gwp = twd * IMG_W_, ghp = tht * IMG_H_;
                s_px[slot] = cx; s_py[slot] = cy; s_r[slot] = cx * cx + cy * cy;
                s_g[0][slot] = cx - gwp * 0.5f; s_g[1][slot] = cy - ghp * 0.5f;
                s_g[2][slot] = cx + gwp * 0.5f; s_g[3][slot] = cy + ghp * 0.5f;
                float c0 = sigmoidf_(raw[rbase + 1 * (size_t)HW_ + p]);
                float c1 = sigmoidf_(raw[rbase + 2 * (size_t)HW_ + p]);
                float wsg = sigmoidf_(raw[rbase + 3 * (size_t)HW_ + p]);
                float hsg = sigmoidf_(raw[rbase + 4 * (size_t)HW_ + p]);
                float pcx = ((float)w + c0) * STRIDE_, pcy = ((float)h + c1) * STRIDE_;
                float pwp = wsg * IMG_W_, php = hsg * IMG_H_;
                s_m[0][slot] = pcx - pwp * 0.5f; s_m[1][slot] = pcy - php * 0.5f;
                s_m[2][slot] = pcx + pwp * 0.5f; s_m[3][slot] = pcy + php * 0.5f;
            }
        }
    }
    __syncthreads();
    const int npos = min(s_cnt, KPOS);
    for (int i = npos + tid; i < KPOS; i += 256) {
        s_px[i] = 0.0f; s_py[i] = 0.0f; s_r[i] = 0.0f;
        s_g[0][i] = s_g[1][i] = s_g[2][i] = s_g[3][i] = 0.0f;
        s_m[0][i] = s_m[1][i] = s_m[2][i] = s_m[3][i] = 0.0f;
    }
    __syncthreads();

    const int wave = tid >> 5;   // tile-row ti
    const int lane = tid & 31;
    const int hi = lane >> 4;    // K/M half select
    const int lo = lane & 15;

    // A operand (32-bit A 16x4 layout: lanes0-15 hold K0/K1, lanes16-31 K2/K3)
    const int ai = 16 * wave + lo;
    v2f A;
    if (hi == 0) { A.x = -2.0f * s_px[ai]; A.y = -2.0f * s_py[ai]; }
    else         { A.x = s_r[ai];          A.y = 1.0f; }

    float bestd[8]; int bestj[8];
#pragma unroll
    for (int v = 0; v < 8; ++v) { bestd[v] = 3.0e30f; bestj[v] = 0; }

#pragma unroll
    for (int tj = 0; tj < 8; ++tj) {
        const int bj = 16 * tj + lo;
        v2f Bm;
        if (hi == 0) { Bm.x = s_px[bj]; Bm.y = s_py[bj]; }
        else         { Bm.x = 1.0f;     Bm.y = s_r[bj]; }
        v8f C = {0.f, 0.f, 0.f, 0.f, 0.f, 0.f, 0.f, 0.f};
        v8f D;
#if __has_builtin(__builtin_amdgcn_wmma_f32_16x16x4_f32)
        D = __builtin_amdgcn_wmma_f32_16x16x4_f32(false, A, false, Bm,
                                                  (short)0, C, false, false);
#else
        // VALU fallback matching the WMMA C/D layout (row = 16*wave+8*hi+v)
#pragma unroll
        for (int v = 0; v < 8; ++v) {
            int gi = 16 * wave + 8 * hi + v;
            float dx = s_px[gi] - s_px[bj];
            float dy = s_py[gi] - s_py[bj];
            D[v] = dx * dx + dy * dy;
        }
#endif
#pragma unroll
        for (int v = 0; v < 8; ++v) {
            const int gi = 16 * wave + 8 * hi + v;
            int jj = 16 * tj + lo;
            float d = D[v];
            if (jj == gi || jj >= npos) d = 1.0e30f;  // diag + invalid cols -> BIG
#pragma unroll
            for (int off = 1; off < 16; off <<= 1) {  // min over 16 cols
                float od = __shfl_xor(d, off, 32);
                int   oj = __shfl_xor(jj, off, 32);
                if (od < d) { d = od; jj = oj; }
            }
            if (d < bestd[v]) { bestd[v] = d; bestj[v] = jj; }
        }
    }
    if (lo == 0) {
#pragma unroll
        for (int v = 0; v < 8; ++v) s_nn[16 * wave + 8 * hi + v] = bestj[v];
    }
    __syncthreads();

    // IoA of pred box vs NN's gt box, mean over valid rows
    float ioa = 0.0f;
    if (tid < KPOS && tid < npos) {
        const int nn = s_nn[tid];
        float nx1 = s_g[0][nn], ny1 = s_g[1][nn], nx2 = s_g[2][nn], ny2 = s_g[3][nn];
        float mx1 = s_m[0][tid], my1 = s_m[1][tid], mx2 = s_m[2][tid], my2 = s_m[3][tid];
        float iw = fmaxf(fminf(mx2, nx2) - fmaxf(mx1, nx1), 0.0f);
        float ih = fmaxf(fminf(my2, ny2) - fmaxf(my1, ny1), 0.0f);
        float na = fmaxf(nx2 - nx1, 0.0f) * fmaxf(ny2 - ny1, 0.0f) + 1e-7f;
        ioa = iw * ih / na;
    }
    if (tid < KPOS) s_red[tid] = ioa;
    __syncthreads();
    if (tid == 0) {
        float sum = 0.0f;
        for (int i = 0; i < KPOS; ++i) sum += s_red[i];
        float mean = sum / fmaxf((float)npos, 1.0f);
        ws[72 + b] = (float)npos;              // gt_count per image (exact)
        if (npos >= 2) {
            atomicAdd(&ws[3], mean);           // rep numerator
            atomicAdd(&ws[6], 1.0f);           // rep count
        }
    }
}

// ---------------------------------------------------------------------------
// Kernel 3: combine partials into the final scalar.
// ---------------------------------------------------------------------------
__global__ void combine_kernel(const float* __restrict__ ws, float* __restrict__ out)
{
    if (threadIdx.x == 0 && blockIdx.x == 0) {
        float npos_hm = fmaxf(ws[2], 1.0f);
        float hm_l = (ws[0] + ws[1]) / npos_hm;
        float npos = 0.0f, cnt = 0.0f;
        for (int b = 0; b < B_; ++b) {
            npos += ws[72 + b];
            cnt  += fabsf(ws[8 + b] - ws[72 + b]);
        }
        npos = fmaxf(npos, 1.0f);
        float cxy_l = ws[4] / npos;
        float wh_l  = ws[5] / npos;
        float rep_l = ws[3] / fmaxf(ws[6], 1.0f);
        float cnt_l = cnt / (float)B_;
        out[0] = 1.0f * hm_l + 1.0f * cxy_l + 5.0f * wh_l + 0.5f * rep_l + 0.1f * cnt_l;
    }
}

extern "C" void kernel_launch(void* const* d_in, const int* in_sizes, int n_in,
                              void* d_out, int out_size, void* d_ws, size_t ws_size,
                              hipStream_t stream) {
    const float* raw     = (const float*)d_in[0];
    const float* tgt_hm  = (const float*)d_in[1];
    const float* tgt_cxy = (const float*)d_in[2];
    const float* tgt_wh  = (const float*)d_in[3];
    const float* tgt_pos = (const float*)d_in[4];
    float* ws  = (float*)d_ws;
    float* out = (float*)d_out;

    hipMemsetAsync(ws, 0, 136 * sizeof(float), stream);

    dim3 blk(32, 8, 1), grd(W_ / 32, H_ / 8, B_);
    fused_loss_kernel<<<grd, blk, 0, stream>>>(raw, tgt_hm, tgt_cxy, tgt_wh, tgt_pos, ws);
    repulsion_kernel<<<B_, 256, 0, stream>>>(raw, tgt_cxy, tgt_wh, tgt_pos, ws);
    combine_kernel<<<1, 64, 0, stream>>>(ws, out);
}